// Conv2dLocal_89481348645682
// MI455X (gfx1250) — compile-verified
//
#include <hip/hip_runtime.h>

typedef float v2f __attribute__((ext_vector_type(2)));
typedef float v8f __attribute__((ext_vector_type(8)));
typedef int   v4i __attribute__((ext_vector_type(4)));

#define OH_ 30
#define OW_ 30
#define NB_ 64    // batch
#define CIN_ 32
#define H_ 32
#define W_ 32
#define COUT_ 64
#define K_ 288        // CIN*3*3
#define LDK_ 292      // padded LDS row stride in floats (292 % 64 = 36 -> conflict-free)
#define ROWBYTES_ 1168  // LDK_*4, = 73 * 16 (16B aligned rows)

#define AS1 __attribute__((address_space(1)))
#define AS3 __attribute__((address_space(3)))

__global__ __launch_bounds__(128) void Conv2dLocal_wmma_f32(
    const float* __restrict__ x,       // [64,32,32,32]
    const float* __restrict__ weight,  // [30,30,64,32,3,3]
    const float* __restrict__ bias,    // [64,30,30]
    float* __restrict__ out)           // [64,64,30,30]
{
    __shared__ float As[NB_ * LDK_];    // im2col patches: [b][k]
    __shared__ float Bs[COUT_ * LDK_];  // per-location weights: [o][k]

    const int hw = blockIdx.x;          // 0..899
    const int h  = hw / OW_;
    const int w  = hw - h * OW_;
    const int tid = threadIdx.x;

    // ---- stage weights: contiguous [64][288] slab for this location.
    // CDNA5 async global->LDS DMA (ASYNCcnt path): no VGPR round-trip,
    // overlaps with the im2col gather below.
    const float* wblk = weight + (size_t)hw * (COUT_ * K_);
#if __has_builtin(__builtin_amdgcn_global_load_async_to_lds_b128)
    {
        char* gsrc = (char*)wblk;
        // 73728 bytes = 4608 chunks of 16B; LDS row stride padded to 1168B.
        for (int i = tid; i < (COUT_ * K_) / 4; i += 128) {
            int row = i / 72;            // 72 x 16B chunks per 288-float row
            int col = i - row * 72;
            __builtin_amdgcn_global_load_async_to_lds_b128(
                (AS1 v4i*)(gsrc + (size_t)i * 16),
                (AS3 v4i*)((char*)Bs + row * ROWBYTES_ + col * 16),
                /*offset=*/0, /*cpol=*/0);
        }
    }
#else
    for (int i = tid; i < COUT_ * K_; i += 128) {
        int o = i / K_;
        int k = i - o * K_;
        Bs[o * LDK_ + k] = wblk[i];
    }
#endif

    // ---- stage activations (overlaps with async weight DMA):
    // im2col gather, k = c*9 + kh*3 + kw
    for (int i = tid; i < NB_ * K_; i += 128) {
        int b  = i / K_;
        int k  = i - b * K_;
        int c  = k / 9;
        int r  = k - c * 9;
        int kh = r / 3;
        int kw = r - kh * 3;
        As[b * LDK_ + k] = x[((b * CIN_ + c) * H_ + (h + kh)) * W_ + (w + kw)];
    }

#if __has_builtin(__builtin_amdgcn_global_load_async_to_lds_b128)
#if __has_builtin(__builtin_amdgcn_s_wait_asynccnt)
    __builtin_amdgcn_s_wait_asynccnt(0);
#else
    asm volatile("s_wait_asynccnt 0" ::: "memory");
#endif
#endif
    __syncthreads();

    const int lane  = tid & 31;
    const int wv    = tid >> 5;     // wave id 0..3 -> M tile
    const int hi    = lane >> 4;    // 0: K pair {k0,k0+1}, 1: {k0+2,k0+3}
    const int l16   = lane & 15;
    const int arow  = wv * 16 + l16;

    v8f acc0 = {}, acc1 = {}, acc2 = {}, acc3 = {};

    for (int k0 = 0; k0 < K_; k0 += 4) {
        const int kk = k0 + hi * 2;
        v2f a  = *(const v2f*)&As[arow * LDK_ + kk];
        v2f b0 = *(const v2f*)&Bs[( 0 + l16) * LDK_ + kk];
        v2f b1 = *(const v2f*)&Bs[(16 + l16) * LDK_ + kk];
        v2f b2 = *(const v2f*)&Bs[(32 + l16) * LDK_ + kk];
        v2f b3 = *(const v2f*)&Bs[(48 + l16) * LDK_ + kk];
        acc0 = __builtin_amdgcn_wmma_f32_16x16x4_f32(false, a, false, b0, (short)0, acc0, false, false);
        acc1 = __builtin_amdgcn_wmma_f32_16x16x4_f32(false, a, false, b1, (short)0, acc1, false, false);
        acc2 = __builtin_amdgcn_wmma_f32_16x16x4_f32(false, a, false, b2, (short)0, acc2, false, false);
        acc3 = __builtin_amdgcn_wmma_f32_16x16x4_f32(false, a, false, b3, (short)0, acc3, false, false);
    }

    // ---- epilogue: D element (vgpr r, lane l) -> M = r + 8*(l>=16), N = l%16
    // Output is write-once: use non-temporal stores to avoid L2 pollution.
    const int Mbase = wv * 16 + hi * 8;
    v8f accs[4] = {acc0, acc1, acc2, acc3};
    #pragma unroll
    for (int n = 0; n < 4; ++n) {
        const int o  = n * 16 + l16;
        const float bv = bias[(o * OH_ + h) * OW_ + w];
        #pragma unroll
        for (int r = 0; r < 8; ++r) {
            const int b = Mbase + r;
            __builtin_nontemporal_store(
                accs[n][r] + bv,
                &out[(((size_t)b * COUT_ + o) * OH_ + h) * OW_ + w]);
        }
    }
}

extern "C" void kernel_launch(void* const* d_in, const int* in_sizes, int n_in,
                              void* d_out, int out_size, void* d_ws, size_t ws_size,
                              hipStream_t stream) {
    const float* x      = (const float*)d_in[0];
    const float* weight = (const float*)d_in[1];
    const float* bias   = (const float*)d_in[2];
    float* out          = (float*)d_out;

    dim3 grid(OH_ * OW_);   // 900 spatial locations
    dim3 block(128);        // 4 wave32 waves
    Conv2dLocal_wmma_f32<<<grid, block, 0, stream>>>(x, weight, bias, out);
}